// GraphConvBlock_4604204941839
// MI455X (gfx1250) — compile-verified
//
#include <hip/hip_runtime.h>
#include <cstdint>

#define C 128           // C_IN == C_OUT == 128
#define LEAKY 0.01f
#define BN_EPS 1e-5f

typedef float v2f __attribute__((ext_vector_type(2)));
typedef float v8f __attribute__((ext_vector_type(8)));

// ---------------- init: deg = 1 (self loop), stats = 0 ----------------
__global__ __launch_bounds__(256) void k_init(unsigned* __restrict__ deg,
                                              float* __restrict__ stats, int N) {
    int i = blockIdx.x * 256 + threadIdx.x;
    if (i < N) deg[i] = 1u;
    if (blockIdx.x == 0) stats[threadIdx.x] = 0.0f;   // 256 floats: sum/sumsq
}

// ---------------- degree accumulation over edge dst ----------------
__global__ __launch_bounds__(256) void k_deg(const int* __restrict__ dst,
                                             unsigned* __restrict__ deg, int E) {
    int e = blockIdx.x * 256 + threadIdx.x;
    if (e < E) atomicAdd(&deg[dst[e]], 1u);
}

// ---------------- dinv = rsqrt(deg), in place (uint -> float) ----------------
__global__ __launch_bounds__(256) void k_dinv(float* __restrict__ dinv, int N) {
    int i = blockIdx.x * 256 + threadIdx.x;
    if (i < N) {
        unsigned d = reinterpret_cast<unsigned*>(dinv)[i];
        dinv[i] = rsqrtf((float)d);   // d >= 1 always (self loop)
    }
}

// ---------------- h = x @ W via V_WMMA_F32_16X16X4_F32 ----------------
// Block = 32 rows x 128 cols. 8 waves: wave>>2 picks row-half (16 rows),
// wave&3 picks a 32-col pair -> TWO independent 16x16 accumulators per wave
// (halves A traffic per WMMA, gives 2 independent WMMA chains).
// Epilogue fuses the self-loop term: out = h*dinv[row]^2 + bias  (saves a
// full 102MB pass that k_out_init used to do).
// A layout (16x4 f32): lane L<16: M=L, K={0,1}; lane 16+L: M=L, K={2,3}
// B layout (4x16 f32): lane L<16: N=L, K={0,1}; lane 16+L: N=L, K={2,3}
// C/D (16x16 f32, 8 VGPRs): vgpr r -> M=r (lanes 0-15), M=r+8 (lanes 16-31)
__global__ __launch_bounds__(256) void k_gemm(const float* __restrict__ x,
                                              const float* __restrict__ W,
                                              const float* __restrict__ dinv,
                                              const float* __restrict__ bias,
                                              float* __restrict__ h,
                                              float* __restrict__ out) {
    __shared__ float Xs[32 * 132];                     // padded: conflict-free reads
    const int tid = threadIdx.x;
    const int rowBase = blockIdx.x * 32;

    // stage 32 rows of x into LDS (4096 floats, 16 per thread, coalesced)
    for (int i = tid; i < 32 * C; i += 256) {
        int m = i >> 7, k = i & (C - 1);
        Xs[m * 132 + k] = x[(size_t)(rowBase + m) * C + k];
    }
    __syncthreads();

    const int wave    = tid >> 5;
    const int lane    = tid & 31;
    const int rowHalf = wave >> 2;                     // 0 or 1
    const int colBase = (wave & 3) * 32;               // two 16-col tiles
    const int m  = lane & 15;
    const int kk = (lane >> 4) << 1;                   // 0 or 2
    const float* Xrow = Xs + (rowHalf * 16 + m) * 132;

    v8f acc0 = {0.f, 0.f, 0.f, 0.f, 0.f, 0.f, 0.f, 0.f};
    v8f acc1 = {0.f, 0.f, 0.f, 0.f, 0.f, 0.f, 0.f, 0.f};
    const float* wp = W + colBase + m;                 // W columns for this lane

    #pragma unroll 4
    for (int k = 0; k < C; k += 4) {
        v2f a, b0, b1;
        a.x  = Xrow[k + kk];
        a.y  = Xrow[k + kk + 1];
        b0.x = wp[(size_t)(k + kk) * C];
        b0.y = wp[(size_t)(k + kk + 1) * C];
        b1.x = wp[(size_t)(k + kk) * C + 16];
        b1.y = wp[(size_t)(k + kk + 1) * C + 16];
        acc0 = __builtin_amdgcn_wmma_f32_16x16x4_f32(
            false, a, false, b0, (short)0, acc0, false, false);
        acc1 = __builtin_amdgcn_wmma_f32_16x16x4_f32(
            false, a, false, b1, (short)0, acc1, false, false);
    }

    const int mOff = (lane >> 4) * 8;
    const int col0 = colBase + m;
    const float bv0 = bias[col0];
    const float bv1 = bias[col0 + 16];
    const int row0 = rowBase + rowHalf * 16 + mOff;
    #pragma unroll
    for (int r = 0; r < 8; ++r) {
        const int row = row0 + r;
        float s = dinv[row]; s = s * s;                // self-loop norm
        const size_t o = (size_t)row * C + col0;
        h[o]        = acc0[r];
        h[o + 16]   = acc1[r];
        out[o]      = fmaf(acc0[r], s, bv0);
        out[o + 16] = fmaf(acc1[r], s, bv1);
    }
}

// ---------------- edge scatter-add: one wave32 per edge ----------------
__global__ __launch_bounds__(256) void k_edge_agg(const int* __restrict__ src,
                                                  const int* __restrict__ dst,
                                                  const float* __restrict__ h,
                                                  const float* __restrict__ dinv,
                                                  float* __restrict__ out, int E) {
    int gw = (blockIdx.x * 256 + threadIdx.x) >> 5;    // global wave id == edge id
    int lane = threadIdx.x & 31;
    if (gw >= E) return;
    int s = src[gw], d = dst[gw];
    const float* hrow = h + (size_t)s * C;
    __builtin_prefetch(hrow + lane * 4, 0, 0);         // global_prefetch_b8
    float norm = dinv[s] * dinv[d];
    float* orow = out + (size_t)d * C;
    #pragma unroll
    for (int j = 0; j < 4; ++j) {
        int c = lane + 32 * j;                          // coalesced 128B per step
        unsafeAtomicAdd(&orow[c], hrow[c] * norm);      // native global_atomic_add_f32
    }
}

// ---------------- LeakyReLU in place + per-channel sum/sumsq ----------------
__global__ __launch_bounds__(256) void k_act_stats(float* __restrict__ out,
                                                   float* __restrict__ stats, int N) {
    __shared__ float ls[512];
    const int c = threadIdx.x & (C - 1);
    const int r = threadIdx.x >> 7;                    // 0 or 1
    const int base = blockIdx.x * 64;
    float s = 0.f, q = 0.f;
    #pragma unroll 4
    for (int i = 0; i < 32; ++i) {
        int n = base + r + 2 * i;
        if (n < N) {
            float v = out[(size_t)n * C + c];
            v = v > 0.f ? v : v * LEAKY;
            out[(size_t)n * C + c] = v;
            s += v; q += v * v;
        }
    }
    ls[threadIdx.x] = s;
    ls[256 + threadIdx.x] = q;
    __syncthreads();
    if (r == 0) {
        unsafeAtomicAdd(&stats[c],     ls[c] + ls[c + 128]);
        unsafeAtomicAdd(&stats[C + c], ls[256 + c] + ls[256 + c + 128]);
    }
}

// ---------------- stats -> (scale, shift), in place ----------------
__global__ void k_bn_prep(float* __restrict__ stats,
                          const float* __restrict__ gamma,
                          const float* __restrict__ beta, int N) {
    int c = threadIdx.x;                               // 128 threads
    float invN = 1.0f / (float)N;
    float mean = stats[c] * invN;
    float var  = stats[C + c] * invN - mean * mean;    // biased variance
    float scale = gamma[c] * rsqrtf(var + BN_EPS);
    float shift = beta[c] - mean * scale;
    stats[c] = scale;
    stats[C + c] = shift;
}

// ---------------- out = out * scale[c] + shift[c] ----------------
__global__ __launch_bounds__(256) void k_affine(float* __restrict__ out,
                                                const float* __restrict__ stats,
                                                long total) {
    long idx = ((long)blockIdx.x * 256 + threadIdx.x) * 4;
    if (idx < total) {
        int c = (int)(idx & (C - 1));
        float4 v = *(float4*)(out + idx);
        v.x = fmaf(v.x, stats[c],     stats[C + c]);
        v.y = fmaf(v.y, stats[c + 1], stats[C + c + 1]);
        v.z = fmaf(v.z, stats[c + 2], stats[C + c + 2]);
        v.w = fmaf(v.w, stats[c + 3], stats[C + c + 3]);
        *(float4*)(out + idx) = v;
    }
}

extern "C" void kernel_launch(void* const* d_in, const int* in_sizes, int n_in,
                              void* d_out, int out_size, void* d_ws, size_t ws_size,
                              hipStream_t stream) {
    const float* x     = (const float*)d_in[0];
    const int*   edges = (const int*)  d_in[1];        // [2, E] flat: src then dst
    const float* W     = (const float*)d_in[2];
    const float* bias  = (const float*)d_in[3];
    const float* gamma = (const float*)d_in[4];
    const float* beta  = (const float*)d_in[5];

    const int N = in_sizes[0] / C;                     // 100000
    const int E = in_sizes[1] / 2;                     // 1600000
    const long total = (long)N * C;

    const int* srcArr = edges;
    const int* dstArr = edges + E;

    float* h     = (float*)d_ws;                       // N*C floats
    float* dinv  = h + total;                          // N floats (deg as u32 first)
    float* stats = dinv + N;                           // 256 floats
    float* out   = (float*)d_out;

    int gN   = (N + 255) / 256;
    int gE   = (E + 255) / 256;
    int gV4  = (int)((total / 4 + 255) / 256);
    int gEW  = (E + 7) / 8;                            // 8 waves (edges) per block
    int gST  = (N + 63) / 64;

    k_init     <<<gN,   256, 0, stream>>>((unsigned*)dinv, stats, N);
    k_deg      <<<gE,   256, 0, stream>>>(dstArr, (unsigned*)dinv, E);
    k_dinv     <<<gN,   256, 0, stream>>>(dinv, N);
    k_gemm     <<<N/32, 256, 0, stream>>>(x, W, dinv, bias, h, out);
    k_edge_agg <<<gEW,  256, 0, stream>>>(srcArr, dstArr, h, dinv, out, E);
    k_act_stats<<<gST,  256, 0, stream>>>(out, stats, N);
    k_bn_prep  <<<1,    128, 0, stream>>>(stats, gamma, beta, N);
    k_affine   <<<gV4,  256, 0, stream>>>(out, stats, total);
}